// GNN_3951369912442
// MI455X (gfx1250) — compile-verified
//
#include <hip/hip_runtime.h>
#include <hip/hip_bf16.h>
#include <math.h>

typedef __attribute__((ext_vector_type(2))) float v2f;
typedef __attribute__((ext_vector_type(8))) float v8f;

static inline int cdiv(long long a, long long b) { return (int)((a + b - 1) / b); }

// ---------------------------------------------------------------- utilities
__global__ void k_fill1(float* __restrict__ p, int n) {
    int i = blockIdx.x * blockDim.x + threadIdx.x;
    if (i < n) p[i] = 1.0f;
}

__global__ void k_deg(const int* __restrict__ col, float* __restrict__ deg, int E) {
    int e = blockIdx.x * blockDim.x + threadIdx.x;
    if (e < E) atomicAdd(&deg[col[e]], 1.0f);
}

__global__ void k_rsqrt(float* __restrict__ d, int n) {
    int i = blockIdx.x * blockDim.x + threadIdx.x;
    if (i < n) d[i] = rsqrtf(d[i]);   // deg >= 1 always (self loops)
}

// ------------------------------------------------ WMMA: xw = x[N,3] @ W1[3,16]
// One wave -> 16-node x 16-feature tile, K padded 3 -> 4, single v_wmma_f32_16x16x4_f32.
__global__ void k_xw1_wmma(const float* __restrict__ x, const float* __restrict__ W1,
                           float* __restrict__ xw, int n_nodes) {
    int wave = (blockIdx.x * blockDim.x + threadIdx.x) >> 5;
    int lane = threadIdx.x & 31;
    int base = wave << 4;
    if (base >= n_nodes) return;          // wave-uniform: EXEC stays all-ones
    int m  = lane & 15;                   // A: row  / B,C,D: column
    int kh = lane >> 4;                   // K-half select
    int k0 = 2 * kh;                      // K in {0,1} or {2,3}
    const float* xr = x + (size_t)(base + m) * 3;
    v2f a, b;
    a.x = xr[k0];                                     // k=0 or k=2 (valid)
    a.y = (k0 + 1 < 3) ? xr[k0 + 1] : 0.0f;           // k=3 is zero pad
    b.x = W1[k0 * 16 + m];
    b.y = (k0 + 1 < 3) ? W1[(k0 + 1) * 16 + m] : 0.0f;
    v8f c = {};
    c = __builtin_amdgcn_wmma_f32_16x16x4_f32(false, a, false, b, (short)0, c, false, false);
    float* o = xw + (size_t)base * 16;
    int rb = kh * 8;
#pragma unroll
    for (int i = 0; i < 8; ++i) o[(size_t)(rb + i) * 16 + m] = c[i];
}

// ------------------------------------------------ WMMA: out = h[N,16] @ W[16,16]
// One wave -> 16x16 tile, 4 chained v_wmma_f32_16x16x4_f32 over K slices.
__global__ void k_h_w_wmma(const float* __restrict__ h, const float* __restrict__ W,
                           float* __restrict__ out, int n_nodes) {
    int wave = (blockIdx.x * blockDim.x + threadIdx.x) >> 5;
    int lane = threadIdx.x & 31;
    int base = wave << 4;
    if (base >= n_nodes) return;          // wave-uniform guard
    int m  = lane & 15;
    int kh = lane >> 4;
    const float* hr = h + (size_t)(base + m) * 16;
    v8f c = {};
#pragma unroll
    for (int s = 0; s < 4; ++s) {
        int k0 = 4 * s + 2 * kh;
        v2f a, b;
        a.x = hr[k0];
        a.y = hr[k0 + 1];
        b.x = W[k0 * 16 + m];
        b.y = W[(k0 + 1) * 16 + m];
        c = __builtin_amdgcn_wmma_f32_16x16x4_f32(false, a, false, b, (short)0, c, false, false);
    }
    float* o = out + (size_t)base * 16;
    int rb = kh * 8;
#pragma unroll
    for (int i = 0; i < 8; ++i) o[(size_t)(rb + i) * 16 + m] = c[i];
}

// --------------------------------------------- edge scatter: dst[col] += src[row]*norm
// 16 consecutive lanes = 16 features of one edge -> coalesced loads & atomics.
__global__ void k_edge_scatter(const int* __restrict__ row, const int* __restrict__ col,
                               const float* __restrict__ dis, const float* __restrict__ src,
                               float* __restrict__ dst, int E) {
    int t = blockIdx.x * blockDim.x + threadIdx.x;   // E*16 = 51.2M < 2^31
    int e = t >> 4;
    int k = t & 15;
    if (e >= E) return;
    int r = row[e];
    int c = col[e];
    float nm = dis[r] * dis[c];
    atomicAdd(&dst[(size_t)c * 16 + k], src[(size_t)r * 16 + k] * nm);
}

// self loop: dst[i] += src[i] * dis[i]^2   (one owner per location, no atomics needed)
__global__ void k_self_loop(const float* __restrict__ src, const float* __restrict__ dis,
                            float* __restrict__ dst, int n16) {
    int t = blockIdx.x * blockDim.x + threadIdx.x;
    if (t >= n16) return;
    float d = dis[t >> 4];
    dst[t] += src[t] * d * d;
}

__global__ void k_bias_relu(float* __restrict__ h, const float* __restrict__ b, int n16) {
    int t = blockIdx.x * blockDim.x + threadIdx.x;
    if (t >= n16) return;
    h[t] = fmaxf(h[t] + b[t & 15], 0.0f);
}

// pooling: g[batch[i]] += h2[i] + b2  (bias folded per node, matches reference)
__global__ void k_pool(const float* __restrict__ h2, const float* __restrict__ b2,
                       const int* __restrict__ batch, float* __restrict__ g, int n16) {
    int t = blockIdx.x * blockDim.x + threadIdx.x;
    if (t >= n16) return;
    int i = t >> 4;
    int k = t & 15;
    atomicAdd(&g[(size_t)batch[i] * 16 + k], h2[t] + b2[k]);
}

// head: logits = g @ Wl + bl ; log_softmax over 7 classes. One thread per graph.
__global__ void k_head(const float* __restrict__ g, const float* __restrict__ Wl,
                       const float* __restrict__ bl, float* __restrict__ out, int G) {
    int i = blockIdx.x * blockDim.x + threadIdx.x;
    if (i >= G) return;
    float gv[16];
#pragma unroll
    for (int k = 0; k < 16; ++k) gv[k] = g[(size_t)i * 16 + k];
    float lg[7];
#pragma unroll
    for (int j = 0; j < 7; ++j) {
        float acc = bl[j];
#pragma unroll
        for (int k = 0; k < 16; ++k) acc += gv[k] * Wl[k * 7 + j];
        lg[j] = acc;
    }
    float mx = lg[0];
#pragma unroll
    for (int j = 1; j < 7; ++j) mx = fmaxf(mx, lg[j]);
    float s = 0.0f;
#pragma unroll
    for (int j = 0; j < 7; ++j) s += __expf(lg[j] - mx);
    float lse = mx + __logf(s);
#pragma unroll
    for (int j = 0; j < 7; ++j) out[(size_t)i * 7 + j] = lg[j] - lse;
}

// ---------------------------------------------------------------- launcher
extern "C" void kernel_launch(void* const* d_in, const int* in_sizes, int n_in,
                              void* d_out, int out_size, void* d_ws, size_t ws_size,
                              hipStream_t stream) {
    const float* x     = (const float*)d_in[0];
    const int*   ei    = (const int*)d_in[1];   // [2,E]: row then col
    /* d_in[2] = edge_attr : unused by the reference */
    const int*   batch = (const int*)d_in[3];
    const float* W1    = (const float*)d_in[4];
    const float* b1    = (const float*)d_in[5];
    const float* W2    = (const float*)d_in[6];
    const float* b2    = (const float*)d_in[7];
    const float* Wl    = (const float*)d_in[8];
    const float* bl    = (const float*)d_in[9];
    float*       out   = (float*)d_out;

    const int N = in_sizes[0] / 3;   // 100000
    const int E = in_sizes[1] / 2;   // 3200000
    const int G = out_size / 7;      // 1000

    // workspace layout (floats)
    float* dis = (float*)d_ws;                 // N
    float* xw  = dis + N;                      // N*16  (reused as hw2)
    float* h1  = xw + (size_t)N * 16;          // N*16
    float* h2  = h1 + (size_t)N * 16;          // N*16
    float* g   = h2 + (size_t)N * 16;          // G*16

    const int* row = ei;
    const int* col = ei + E;

    const int B = 256;
    const int n16 = N * 16;
    const int tiles = cdiv(N, 16);
    const int wmma_blocks = cdiv((long long)tiles * 32, B);

    // zero h1, h2, g (contiguous region)
    hipMemsetAsync(h1, 0, ((size_t)2 * n16 + (size_t)G * 16) * sizeof(float), stream);

    // degree + symmetric normalization
    k_fill1<<<cdiv(N, B), B, 0, stream>>>(dis, N);
    k_deg<<<cdiv(E, B), B, 0, stream>>>(col, dis, E);
    k_rsqrt<<<cdiv(N, B), B, 0, stream>>>(dis, N);

    // conv1: xw = x @ W1 (WMMA), scatter, self-loop, bias+relu
    k_xw1_wmma<<<wmma_blocks, B, 0, stream>>>(x, W1, xw, N);
    k_edge_scatter<<<cdiv((long long)E * 16, B), B, 0, stream>>>(row, col, dis, xw, h1, E);
    k_self_loop<<<cdiv(n16, B), B, 0, stream>>>(xw, dis, h1, n16);
    k_bias_relu<<<cdiv(n16, B), B, 0, stream>>>(h1, b1, n16);

    // conv2: hw2 = h1 @ W2 (WMMA, reuse xw), scatter, self-loop
    k_h_w_wmma<<<wmma_blocks, B, 0, stream>>>(h1, W2, xw, N);
    k_edge_scatter<<<cdiv((long long)E * 16, B), B, 0, stream>>>(row, col, dis, xw, h2, E);
    k_self_loop<<<cdiv(n16, B), B, 0, stream>>>(xw, dis, h2, n16);

    // pooling (+b2 folded per node) and classifier head with log_softmax
    k_pool<<<cdiv(n16, B), B, 0, stream>>>(h2, b2, batch, g, n16);
    k_head<<<cdiv(G, 128), 128, 0, stream>>>(g, Wl, bl, out, G);
}